// CombinedTemporalFocalBCELoss_46995532152891
// MI455X (gfx1250) — compile-verified
//
#include <hip/hip_runtime.h>
#include <cstdint>

// CombinedTemporalFocalBCELoss for MI455X (gfx1250).
// Memory-bound streaming reduction: 128 MB in / 4 B out -> ~5.6 us floor at
// 23.3 TB/s. Strategy: b128 streaming loads for outputs, gfx1250 async
// global->LDS DMA (ASYNCcnt) to stage the shared target tiles (+halo) once
// per block, wave32 shuffle reduction, deterministic two-pass mean.

#define THREADS 256
#define TILE    2048   // elements per block = 8 per thread

// ---- gfx1250 async global->LDS copy (16 B per active lane) ----------------
__device__ __forceinline__ void async_ld_b128(uint32_t lds_byte_addr, const void* gaddr) {
  // VDST = LDS byte address (low 32 bits of a flat LDS pointer), VADDR = 64-bit
  // global address, saddr = off (GV mode). Tracked by ASYNCcnt.
  asm volatile("global_load_async_to_lds_b128 %0, %1, off"
               :
               : "v"(lds_byte_addr), "v"((uint64_t)(uintptr_t)gaddr)
               : "memory");
}

__device__ __forceinline__ void wait_asynccnt0() {
  asm volatile("s_wait_asynccnt 0" ::: "memory");
}

// ---- per-element loss, matches the JAX reference ---------------------------
__device__ __forceinline__ float elem_loss(float x, int t, int anyprev) {
  const float ALPHA = 0.25f, TW = 0.8f, EPS = 1e-8f;
  float e  = __expf(-fabsf(x));           // exp(-|x|) in (0,1]
  float l1 = __logf(1.0f + e);            // log1p(exp(-|x|))
  float bce = fmaxf(x, 0.0f) - x * (float)t + l1;
  float r  = 1.0f / (1.0f + e);
  float p  = (x >= 0.0f) ? r : e * r;     // sigmoid(x), stable, one exp total
  float pt = t ? p : 1.0f - p;
  float om = 1.0f - pt;
  float focal = -ALPHA * om * om * __logf(pt + EPS);
  float c = 0.5f * bce + 0.5f * focal;
  if (t && (anyprev != 0)) c *= TW;       // temporal reweight
  return c;
}

__device__ __forceinline__ float wave_reduce(float v) {
  #pragma unroll
  for (int off = 16; off > 0; off >>= 1)
    v += __shfl_down(v, off, 32);         // wave32
  return v;
}

// ---- pass 1: per-block partial sums ---------------------------------------
__global__ __launch_bounds__(THREADS)
void loss_partial_kernel(const float* __restrict__ x, const int* __restrict__ t,
                         float* __restrict__ partial, int n) {
  __shared__ int   s_t[TILE + 8];         // [0..7] halo = targets[base-8..base-1]
  __shared__ float s_w[THREADS / 32];
  const int tid = threadIdx.x;
  const long long base = (long long)blockIdx.x * TILE;
  float sum = 0.0f;

  if (base + TILE <= (long long)n) {
    // Async-DMA this tile's targets (8 KB) + halo into LDS.
    async_ld_b128((uint32_t)(uintptr_t)&s_t[8 + 4 * tid],        t + base + 4 * tid);
    async_ld_b128((uint32_t)(uintptr_t)&s_t[8 + 1024 + 4 * tid], t + base + 1024 + 4 * tid);
    if (blockIdx.x == 0) {
      if (tid < 8) s_t[tid] = 0;          // clipped window before element 0
    } else if (tid < 2) {
      async_ld_b128((uint32_t)(uintptr_t)&s_t[4 * tid], t + base - 8 + 4 * tid);
    }
    // Overlap: stream this thread's 8 outputs with b128 loads.
    const float4* xv = (const float4*)(x + base) + 2 * tid;
    float4 xa = xv[0];
    float4 xb = xv[1];
    wait_asynccnt0();
    __syncthreads();

    const int j0 = tid * 8;
    int4 p0 = *(const int4*)&s_t[j0];       // targets[base+j0-8 .. j0-5]
    int4 p1 = *(const int4*)&s_t[j0 + 4];   // targets[base+j0-4 .. j0-1]
    int4 t0 = *(const int4*)&s_t[j0 + 8];   // own targets 0..3
    int4 t1 = *(const int4*)&s_t[j0 + 12];  // own targets 4..7
    int a[16] = {p0.x, p0.y, p0.z, p0.w, p1.x, p1.y, p1.z, p1.w,
                 t0.x, t0.y, t0.z, t0.w, t1.x, t1.y, t1.z, t1.w};
    float xs[8] = {xa.x, xa.y, xa.z, xa.w, xb.x, xb.y, xb.z, xb.w};
    #pragma unroll
    for (int k = 0; k < 8; ++k) {
      int anyprev = a[k + 3] | a[k + 4] | a[k + 5] | a[k + 6] | a[k + 7];
      sum += elem_loss(xs[k], a[k + 8], anyprev);
    }
  } else {
    // Ragged tail (never hit for N = 16M = 8192 full tiles; kept for safety).
    for (int j = tid; j < TILE; j += THREADS) {
      long long i = base + j;
      if (i < (long long)n) {
        int anyprev = 0;
        #pragma unroll
        for (int w = 1; w <= 5; ++w)
          if (i - w >= 0) anyprev |= t[i - w];
        sum += elem_loss(x[i], t[i], anyprev);
      }
    }
  }

  sum = wave_reduce(sum);
  if ((tid & 31) == 0) s_w[tid >> 5] = sum;
  __syncthreads();
  if (tid == 0) {
    float b = 0.0f;
    #pragma unroll
    for (int i = 0; i < THREADS / 32; ++i) b += s_w[i];
    partial[blockIdx.x] = b;
  }
}

// ---- pass 2: deterministic final reduce + mean ----------------------------
__global__ __launch_bounds__(THREADS)
void final_reduce_kernel(const float* __restrict__ partial, int nparts,
                         float* __restrict__ out, float inv_n) {
  __shared__ float s_w[THREADS / 32];
  float sum = 0.0f;
  for (int i = threadIdx.x; i < nparts; i += THREADS) sum += partial[i];
  sum = wave_reduce(sum);
  if ((threadIdx.x & 31) == 0) s_w[threadIdx.x >> 5] = sum;
  __syncthreads();
  if (threadIdx.x == 0) {
    float tot = 0.0f;
    #pragma unroll
    for (int i = 0; i < THREADS / 32; ++i) tot += s_w[i];
    out[0] = tot * inv_n;
  }
}

extern "C" void kernel_launch(void* const* d_in, const int* in_sizes, int n_in,
                              void* d_out, int out_size, void* d_ws, size_t ws_size,
                              hipStream_t stream) {
  const float* x = (const float*)d_in[0];   // outputs, f32
  const int*   t = (const int*)d_in[1];     // targets, i32
  float* out = (float*)d_out;
  const int n = in_sizes[0];
  const int nblocks = (n + TILE - 1) / TILE;      // 8192 for N = 16,777,216
  float* partial = (float*)d_ws;                  // 32 KB of scratch

  loss_partial_kernel<<<nblocks, THREADS, 0, stream>>>(x, t, partial, n);
  final_reduce_kernel<<<1, THREADS, 0, stream>>>(partial, nblocks, out,
                                                 1.0f / (float)n);
}